// GraphSAGELinkPredictor_78254304133410
// MI455X (gfx1250) — compile-verified
//
#include <hip/hip_runtime.h>
#include <hip/hip_bf16.h>

#define NU 100000
#define NR 250000
#define EE 1500000
#define LL 500000

typedef float v2f __attribute__((ext_vector_type(2)));
typedef float v8f __attribute__((ext_vector_type(8)));

static __device__ __forceinline__ v8f wmma4(v2f a, v2f b, v8f c) {
  // V_WMMA_F32_16X16X4_F32 : D = A(16x4) * B(4x16) + C(16x16), fp32 throughout
  return __builtin_amdgcn_wmma_f32_16x16x4_f32(false, a, false, b, (short)0, c,
                                               false, false);
}

// ---------------------------------------------------------------------------
// Y[N x 64] = X[N x K] @ W[K x 64] + bias   (WMMA, W staged in LDS, k-pair
// interleaved so each B fragment is one ds_load_b64 into an even VGPR pair).
// block = 256 threads = 8 waves; each wave owns a 16-row x 64-col output tile.
// ---------------------------------------------------------------------------
template <int K>
__global__ __launch_bounds__(256) void gemm_bias_kernel(
    const float* __restrict__ X, const float* __restrict__ W,
    const float* __restrict__ bias, float* __restrict__ Y, int N) {
  __shared__ float sW[K * 64];  // sW[(k/2)*128 + col*2 + (k&1)] = W[k*64+col]
  __shared__ float sB[64];
  for (int i = threadIdx.x; i < K * 64; i += 256) {
    const int k = i >> 6;
    const int col = i & 63;
    sW[(k >> 1) * 128 + col * 2 + (k & 1)] = W[i];
  }
  if (threadIdx.x < 64) sB[threadIdx.x] = bias[threadIdx.x];
  __syncthreads();

  const int lane = threadIdx.x & 31;
  const int wave = threadIdx.x >> 5;
  const int rlane = lane & 15;
  const int khalf = lane >> 4;
  const int rowBase = blockIdx.x * 128 + wave * 16;
  const int row = rowBase + rlane;
  const int crow = row < N ? row : N - 1;  // clamp: keeps loads branch-free
  const float* xrow = X + (size_t)crow * K;

  v8f acc[4] = {};
  for (int k0 = 0; k0 < K; k0 += 4) {
    const int ka = k0 + khalf * 2;
    const v2f a = *(const v2f*)(xrow + ka);  // global_load_b64
#pragma unroll
    for (int t = 0; t < 4; ++t) {
      const v2f b =
          *(const v2f*)(sW + (ka >> 1) * 128 + (t * 16 + rlane) * 2);
      acc[t] = wmma4(a, b, acc[t]);
    }
  }
#pragma unroll
  for (int t = 0; t < 4; ++t) {
    const int col = t * 16 + rlane;
#pragma unroll
    for (int g = 0; g < 8; ++g) {
      const int orow = rowBase + g + 8 * khalf;
      if (orow < N) Y[(size_t)orow * 64 + col] = acc[t][g] + sB[col];
    }
  }
}

// ---------------------------------------------------------------------------
// Y[N x 64] = (agg/max(cnt,1)) @ Wl + bl + Xdst @ Wr   (dual-accumulate WMMA)
// ---------------------------------------------------------------------------
__global__ __launch_bounds__(256) void sage_linear_kernel(
    const float* __restrict__ agg, const float* __restrict__ cnt,
    const float* __restrict__ xdst, const float* __restrict__ Wl,
    const float* __restrict__ bl, const float* __restrict__ Wr,
    float* __restrict__ Y, int N) {
  __shared__ float sWl[64 * 64];
  __shared__ float sWr[64 * 64];
  __shared__ float sB[64];
  for (int i = threadIdx.x; i < 64 * 64; i += 256) {
    const int k = i >> 6;
    const int col = i & 63;
    const int j = (k >> 1) * 128 + col * 2 + (k & 1);
    sWl[j] = Wl[i];
    sWr[j] = Wr[i];
  }
  if (threadIdx.x < 64) sB[threadIdx.x] = bl[threadIdx.x];
  __syncthreads();

  const int lane = threadIdx.x & 31;
  const int wave = threadIdx.x >> 5;
  const int rlane = lane & 15;
  const int khalf = lane >> 4;
  const int rowBase = blockIdx.x * 128 + wave * 16;
  const int row = rowBase + rlane;
  const int crow = row < N ? row : N - 1;
  const float* arow = agg + (size_t)crow * 64;
  const float* xrow = xdst + (size_t)crow * 64;
  const float inv = 1.0f / fmaxf(cnt[crow], 1.0f);

  v8f acc[4] = {};
  for (int k0 = 0; k0 < 64; k0 += 4) {
    const int ka = k0 + khalf * 2;
    v2f a1 = *(const v2f*)(arow + ka);
    const v2f a2 = *(const v2f*)(xrow + ka);
    a1.x *= inv;
    a1.y *= inv;
#pragma unroll
    for (int t = 0; t < 4; ++t) {
      const int bo = (ka >> 1) * 128 + (t * 16 + rlane) * 2;
      const v2f b1 = *(const v2f*)(sWl + bo);
      const v2f b2 = *(const v2f*)(sWr + bo);
      acc[t] = wmma4(a1, b1, acc[t]);
      acc[t] = wmma4(a2, b2, acc[t]);
    }
  }
#pragma unroll
  for (int t = 0; t < 4; ++t) {
    const int col = t * 16 + rlane;
#pragma unroll
    for (int g = 0; g < 8; ++g) {
      const int orow = rowBase + g + 8 * khalf;
      if (orow < N) Y[(size_t)orow * 64 + col] = acc[t][g] + sB[col];
    }
  }
}

// ---------------------------------------------------------------------------
// warp-per-edge mean aggregation: agg[dst] += xsrc[src]; cnt[dst] += 1
// (a wave reads/atomically-adds one contiguous 256B feature row per edge)
// ---------------------------------------------------------------------------
__global__ __launch_bounds__(256) void scatter_kernel(
    const float* __restrict__ xsrc, const int* __restrict__ src,
    const int* __restrict__ dst, float* __restrict__ agg,
    float* __restrict__ cnt, int nE) {
  const int t = blockIdx.x * blockDim.x + threadIdx.x;
  const int e = t >> 5;
  if (e >= nE) return;
  const int j = t & 31;
  const int s = src[e];
  const int d = dst[e];
  const float2 v = *(const float2*)(xsrc + (size_t)s * 64 + j * 2);
  float* ap = agg + (size_t)d * 64 + j * 2;
  atomicAdd(ap, v.x);
  atomicAdd(ap + 1, v.y);
  if (j == 0) atomicAdd(cnt + d, 1.0f);
}

// ---------------------------------------------------------------------------
// BatchNorm: column sum / sumsq reduce (stats[0..63]=sum, stats[64..127]=sumsq)
// ---------------------------------------------------------------------------
__global__ __launch_bounds__(256) void bn_reduce_kernel(
    const float* __restrict__ Y, int N, float* __restrict__ stats) {
  __shared__ float s1[256];
  __shared__ float s2[256];
  const int col = threadIdx.x & 63;
  const int rg = threadIdx.x >> 6;
  float a = 0.0f, b = 0.0f;
  for (int r = blockIdx.x * 4 + rg; r < N; r += gridDim.x * 4) {
    const float v = Y[(size_t)r * 64 + col];
    a += v;
    b += v * v;
  }
  s1[threadIdx.x] = a;
  s2[threadIdx.x] = b;
  __syncthreads();
  if (threadIdx.x < 64) {
    a = s1[threadIdx.x] + s1[threadIdx.x + 64] + s1[threadIdx.x + 128] +
        s1[threadIdx.x + 192];
    b = s2[threadIdx.x] + s2[threadIdx.x + 64] + s2[threadIdx.x + 128] +
        s2[threadIdx.x + 192];
    atomicAdd(&stats[col], a);
    atomicAdd(&stats[64 + col], b);
  }
}

__global__ __launch_bounds__(256) void bn_apply_kernel(
    float* __restrict__ Y, const float* __restrict__ stats,
    const float* __restrict__ g, const float* __restrict__ b, int N,
    int relu) {
  const int t = blockIdx.x * blockDim.x + threadIdx.x;
  if (t >= N * 64) return;
  const int col = t & 63;
  const float invN = 1.0f / (float)N;
  const float m = stats[col] * invN;
  const float var = stats[64 + col] * invN - m * m;
  const float sc = g[col] * rsqrtf(var + 1e-5f);
  float v = (Y[t] - m) * sc + b[col];
  if (relu) v = fmaxf(v, 0.0f);
  Y[t] = v;
}

// ---------------------------------------------------------------------------
// decoder: warp per label, L2-normalize rows + dot
// ---------------------------------------------------------------------------
__global__ __launch_bounds__(256) void decoder_kernel(
    const float* __restrict__ zu, const float* __restrict__ zr,
    const int* __restrict__ lu, const int* __restrict__ lr,
    float* __restrict__ out, int nL) {
  const int t = blockIdx.x * blockDim.x + threadIdx.x;
  const int i = t >> 5;
  if (i >= nL) return;
  const int lane = t & 31;
  const float2 av = *(const float2*)(zu + (size_t)lu[i] * 64 + lane * 2);
  const float2 cv = *(const float2*)(zr + (size_t)lr[i] * 64 + lane * 2);
  float dot = av.x * cv.x + av.y * cv.y;
  float na = av.x * av.x + av.y * av.y;
  float nc = cv.x * cv.x + cv.y * cv.y;
#pragma unroll
  for (int m = 16; m >= 1; m >>= 1) {
    dot += __shfl_xor(dot, m, 32);
    na += __shfl_xor(na, m, 32);
    nc += __shfl_xor(nc, m, 32);
  }
  if (lane == 0)
    out[i] = dot / (fmaxf(sqrtf(na), 1e-12f) * fmaxf(sqrtf(nc), 1e-12f));
}

// ---------------------------------------------------------------------------
extern "C" void kernel_launch(void* const* d_in, const int* in_sizes, int n_in,
                              void* d_out, int out_size, void* d_ws,
                              size_t ws_size, hipStream_t stream) {
  const float* x_user = (const float*)d_in[0];
  const float* x_recipe = (const float*)d_in[1];
  const float* Wu = (const float*)d_in[2];
  const float* bu = (const float*)d_in[3];
  const float* Wr = (const float*)d_in[4];
  const float* br = (const float*)d_in[5];
  const float* g_in_u = (const float*)d_in[6];
  const float* b_in_u = (const float*)d_in[7];
  const float* g_in_r = (const float*)d_in[8];
  const float* b_in_r = (const float*)d_in[9];
  const float* Wl1_ur = (const float*)d_in[10];
  const float* bl1_ur = (const float*)d_in[11];
  const float* Wr1_ur = (const float*)d_in[12];
  const float* Wl1_ru = (const float*)d_in[13];
  const float* bl1_ru = (const float*)d_in[14];
  const float* Wr1_ru = (const float*)d_in[15];
  const float* g1_u = (const float*)d_in[16];
  const float* b1_u = (const float*)d_in[17];
  const float* g1_r = (const float*)d_in[18];
  const float* b1_r = (const float*)d_in[19];
  const float* Wl2_ur = (const float*)d_in[20];
  const float* bl2_ur = (const float*)d_in[21];
  const float* Wr2_ur = (const float*)d_in[22];
  const float* Wl2_ru = (const float*)d_in[23];
  const float* bl2_ru = (const float*)d_in[24];
  const float* Wr2_ru = (const float*)d_in[25];
  const float* g2_u = (const float*)d_in[26];
  const float* b2_u = (const float*)d_in[27];
  const float* g2_r = (const float*)d_in[28];
  const float* b2_r = (const float*)d_in[29];
  const int* edge_user = (const int*)d_in[30];
  const int* edge_recipe = (const int*)d_in[31];
  const int* label_user = (const int*)d_in[32];
  const int* label_recipe = (const int*)d_in[33];
  float* out = (float*)d_out;

  float* ws = (float*)d_ws;
  const size_t oHU = 0;                       // hu  (later: aggU2)
  const size_t oHR = oHU + (size_t)NU * 64;   // hr  (later: aggR2)
  const size_t oAU = oHR + (size_t)NR * 64;   // aggU (later: zu)
  const size_t oAR = oAU + (size_t)NU * 64;   // aggR (later: zr)
  const size_t oU1 = oAR + (size_t)NR * 64;   // u1
  const size_t oR1 = oU1 + (size_t)NU * 64;   // r1
  const size_t oCU = oR1 + (size_t)NR * 64;   // cntU
  const size_t oCR = oCU + (size_t)NU;        // cntR
  const size_t oST = oCR + (size_t)NR;        // stats (128)

  float* hu = ws + oHU;
  float* hr = ws + oHR;
  float* aggU = ws + oAU;
  float* aggR = ws + oAR;
  float* u1 = ws + oU1;
  float* r1 = ws + oR1;
  float* cntU = ws + oCU;
  float* cntR = ws + oCR;
  float* stats = ws + oST;

  const int RED_BLOCKS = 1024;
  const dim3 blk(256);
  const int gGemmU = (NU + 127) / 128;
  const int gGemmR = (NR + 127) / 128;
  const int gApplyU = (NU * 64 + 255) / 256;
  const int gApplyR = (NR * 64 + 255) / 256;
  const int gScatter = (int)(((size_t)EE * 32 + 255) / 256);
  const int gDec = (int)(((size_t)LL * 32 + 255) / 256);

  // ---- input transforms: hu = relu(BN(x_user@Wu+bu)), hr likewise ----
  gemm_bias_kernel<64><<<gGemmU, blk, 0, stream>>>(x_user, Wu, bu, hu, NU);
  hipMemsetAsync(stats, 0, 128 * sizeof(float), stream);
  bn_reduce_kernel<<<RED_BLOCKS, blk, 0, stream>>>(hu, NU, stats);
  bn_apply_kernel<<<gApplyU, blk, 0, stream>>>(hu, stats, g_in_u, b_in_u, NU, 1);

  gemm_bias_kernel<128><<<gGemmR, blk, 0, stream>>>(x_recipe, Wr, br, hr, NR);
  hipMemsetAsync(stats, 0, 128 * sizeof(float), stream);
  bn_reduce_kernel<<<RED_BLOCKS, blk, 0, stream>>>(hr, NR, stats);
  bn_apply_kernel<<<gApplyR, blk, 0, stream>>>(hr, stats, g_in_r, b_in_r, NR, 1);

  // ---- layer 1 aggregation ----
  hipMemsetAsync(aggU, 0, (size_t)NU * 64 * sizeof(float), stream);
  hipMemsetAsync(aggR, 0, (size_t)NR * 64 * sizeof(float), stream);
  hipMemsetAsync(cntU, 0, (size_t)NU * sizeof(float), stream);
  hipMemsetAsync(cntR, 0, (size_t)NR * sizeof(float), stream);
  scatter_kernel<<<gScatter, blk, 0, stream>>>(hu, edge_user, edge_recipe, aggR, cntR, EE);
  scatter_kernel<<<gScatter, blk, 0, stream>>>(hr, edge_recipe, edge_user, aggU, cntU, EE);

  sage_linear_kernel<<<gGemmR, blk, 0, stream>>>(aggR, cntR, hr, Wl1_ur, bl1_ur, Wr1_ur, r1, NR);
  sage_linear_kernel<<<gGemmU, blk, 0, stream>>>(aggU, cntU, hu, Wl1_ru, bl1_ru, Wr1_ru, u1, NU);

  hipMemsetAsync(stats, 0, 128 * sizeof(float), stream);
  bn_reduce_kernel<<<RED_BLOCKS, blk, 0, stream>>>(u1, NU, stats);
  bn_apply_kernel<<<gApplyU, blk, 0, stream>>>(u1, stats, g1_u, b1_u, NU, 1);
  hipMemsetAsync(stats, 0, 128 * sizeof(float), stream);
  bn_reduce_kernel<<<RED_BLOCKS, blk, 0, stream>>>(r1, NR, stats);
  bn_apply_kernel<<<gApplyR, blk, 0, stream>>>(r1, stats, g1_r, b1_r, NR, 1);

  // ---- layer 2 aggregation (reuse hu/hr buffers as agg, aggU/aggR as z) ----
  float* aggU2 = hu;
  float* aggR2 = hr;
  float* zu = aggU;
  float* zr = aggR;
  hipMemsetAsync(aggU2, 0, (size_t)NU * 64 * sizeof(float), stream);
  hipMemsetAsync(aggR2, 0, (size_t)NR * 64 * sizeof(float), stream);
  hipMemsetAsync(cntU, 0, (size_t)NU * sizeof(float), stream);
  hipMemsetAsync(cntR, 0, (size_t)NR * sizeof(float), stream);
  scatter_kernel<<<gScatter, blk, 0, stream>>>(u1, edge_user, edge_recipe, aggR2, cntR, EE);
  scatter_kernel<<<gScatter, blk, 0, stream>>>(r1, edge_recipe, edge_user, aggU2, cntU, EE);

  sage_linear_kernel<<<gGemmR, blk, 0, stream>>>(aggR2, cntR, r1, Wl2_ur, bl2_ur, Wr2_ur, zr, NR);
  sage_linear_kernel<<<gGemmU, blk, 0, stream>>>(aggU2, cntU, u1, Wl2_ru, bl2_ru, Wr2_ru, zu, NU);

  hipMemsetAsync(stats, 0, 128 * sizeof(float), stream);
  bn_reduce_kernel<<<RED_BLOCKS, blk, 0, stream>>>(zu, NU, stats);
  bn_apply_kernel<<<gApplyU, blk, 0, stream>>>(zu, stats, g2_u, b2_u, NU, 0);
  hipMemsetAsync(stats, 0, 128 * sizeof(float), stream);
  bn_reduce_kernel<<<RED_BLOCKS, blk, 0, stream>>>(zr, NR, stats);
  bn_apply_kernel<<<gApplyR, blk, 0, stream>>>(zr, stats, g2_r, b2_r, NR, 0);

  // ---- decoder ----
  decoder_kernel<<<gDec, blk, 0, stream>>>(zu, zr, label_user, label_recipe, out, LL);

  (void)in_sizes; (void)n_in; (void)out_size; (void)ws_size;
}